// StructuredVariational_75050258530449
// MI455X (gfx1250) — compile-verified
//
#include <hip/hip_runtime.h>

// Problem constants (from reference)
#define DZ     512
#define DY     32
#define NBLK   2048
#define NS     128
#define DTOT   (DZ + NBLK * DY)   // 66048
#define JITTERF 1.0e-4f
#define SROW   516                // padded LDS row stride (floats): 516 % 64 == 4
                                  // -> conflict-free ds_load_b64 A-fragments

typedef float v2f __attribute__((ext_vector_type(2)));
typedef float v8f __attribute__((ext_vector_type(8)));

static __device__ __forceinline__ v8f wmma_k4(v2f a, v2f b, v8f c) {
    // D(16x16,f32) = A(16x4,f32) * B(4x16,f32) + C
    return __builtin_amdgcn_wmma_f32_16x16x4_f32(
        /*neg_a=*/false, a, /*neg_b=*/false, b,
        /*c_mod=*/(short)0, c, /*reuse_a=*/false, /*reuse_b=*/false);
}

static __device__ __forceinline__ v8f v8f_zero() {
    v8f z = {0.f, 0.f, 0.f, 0.f, 0.f, 0.f, 0.f, 0.f};
    return z;
}

// ---------------------------------------------------------------------------
// z_y kernel: out[:, 512+r] = m[512+r] + eps_z @ Lyz^T + blockdiag(eps_y @ Ly^T)
// Grid: 512 blocks x 256 threads (8 waves). Workgroup tile: 128 z_y columns x
// all 128 samples. Wave layout: 2 (M-halves) x 4 (N-quarters); each wave owns
// 4 M-tiles x 2 N-subtiles => 8 D-tiles (64 accum VGPRs), so every A fragment
// feeds 2 WMMAs and every K-step is 4 ds_load_b64 + 2 global_load_b64 + 8 WMMA.
// eps_z staged once in LDS (padded rows, bank-conflict-free).
// A wave's 32 columns == one Ly block, so diag folds in as 8 extra K-steps.
// ---------------------------------------------------------------------------
__global__ __launch_bounds__(256) void zy_wmma_kernel(
    const float* __restrict__ mvec,      // (66048)
    const float* __restrict__ Ly,        // (2048,32,32)
    const float* __restrict__ Lyz,       // (65536,512)
    const float* __restrict__ eps,       // (128,66048)
    float* __restrict__ out)             // (128,66048)
{
    __shared__ float s_eps[NS * SROW];   // 258 KB (<= 320 KB/WGP)

    const int tid = threadIdx.x;

    // Stage eps_z -> LDS with b128 loads/stores (uniform trip count).
    for (int idx = tid; idx < (NS * DZ) / 4; idx += 256) {
        const int s  = idx >> 7;               // 128 float4 per sample row
        const int kk = (idx & 127) << 2;
        const float4 v = *reinterpret_cast<const float4*>(
            eps + (size_t)s * DTOT + kk);
        *reinterpret_cast<float4*>(&s_eps[s * SROW + kk]) = v;
    }
    __syncthreads();

    const int lane = tid & 31;
    const int wave = tid >> 5;
    const int ln   = lane & 15;            // N within subtile / M within tile
    const int hk   = (lane >> 4) << 1;     // K sub-offset: lanes 16-31 -> +2

    const int wm     = wave & 1;           // M half: rows [wm*64, wm*64+64)
    const int wn     = wave >> 1;          // N quarter within the WG tile
    const int n_base = blockIdx.x * 128 + wn * 32;   // 32-aligned => one Ly blk
    const int mrow0  = wm * 64;

    v8f acc[2][4];                         // [n-subtile][m-tile]
    #pragma unroll
    for (int u = 0; u < 2; ++u)
        #pragma unroll
        for (int t = 0; t < 4; ++t) acc[u][t] = v8f_zero();

    // --- main K=512 loop: B streamed from Lyz, A from LDS ------------------
    const float* __restrict__ b0p = Lyz + (size_t)(n_base + ln) * DZ + hk;
    const float* __restrict__ b1p = b0p + (size_t)16 * DZ;
    for (int ks = 0; ks < DZ / 4; ++ks) {
        __builtin_prefetch(b0p + (ks + 32) * 4, 0, 0);   // global_prefetch_b8
        __builtin_prefetch(b1p + (ks + 32) * 4, 0, 0);
        const v2f b0 = *reinterpret_cast<const v2f*>(b0p + ks * 4);
        const v2f b1 = *reinterpret_cast<const v2f*>(b1p + ks * 4);
        const int k = ks * 4 + hk;
        #pragma unroll
        for (int t = 0; t < 4; ++t) {
            const v2f a = *reinterpret_cast<const v2f*>(
                &s_eps[(mrow0 + t * 16 + ln) * SROW + k]);
            acc[0][t] = wmma_k4(a, b0, acc[0][t]);   // v_wmma_f32_16x16x4_f32
            acc[1][t] = wmma_k4(a, b1, acc[1][t]);
        }
    }

    // --- fold block-diagonal term: 8 more K-steps over this wave's block ---
    const int blk = n_base >> 5;
    const float* __restrict__ lyb = Ly + (size_t)blk * (DY * DY);
    const int i0 = ln;                     // block row for N-subtile 0
    const int i1 = ln + 16;                // block row for N-subtile 1
    #pragma unroll
    for (int ks = 0; ks < DY / 4; ++ks) {
        const int j = ks * 4 + hk;
        const v2f r0 = *reinterpret_cast<const v2f*>(lyb + i0 * DY + j);
        const v2f r1 = *reinterpret_cast<const v2f*>(lyb + i1 * DY + j);
        v2f b0, b1;
        b0.x = (j     <= i0) ? r0.x : 0.f; if (j     == i0) b0.x += JITTERF;
        b0.y = (j + 1 <= i0) ? r0.y : 0.f; if (j + 1 == i0) b0.y += JITTERF;
        b1.x = (j     <= i1) ? r1.x : 0.f; if (j     == i1) b1.x += JITTERF;
        b1.y = (j + 1 <= i1) ? r1.y : 0.f; if (j + 1 == i1) b1.y += JITTERF;
        #pragma unroll
        for (int t = 0; t < 4; ++t) {
            const v2f a = *reinterpret_cast<const v2f*>(
                eps + (size_t)(mrow0 + t * 16 + ln) * DTOT + DZ + blk * DY + j);
            acc[0][t] = wmma_k4(a, b0, acc[0][t]);
            acc[1][t] = wmma_k4(a, b1, acc[1][t]);
        }
    }

    // --- add mean, scatter D tiles -----------------------------------------
    const float  mv0   = mvec[DZ + n_base + ln];
    const float  mv1   = mvec[DZ + n_base + 16 + ln];
    const size_t ocol0 = (size_t)DZ + (size_t)(n_base + ln);
    const int    mhi   = (lane >> 4) << 3;       // lanes 16-31 hold rows v+8
    #pragma unroll
    for (int t = 0; t < 4; ++t) {
        #pragma unroll
        for (int v = 0; v < 8; ++v) {
            const int mr = mrow0 + t * 16 + v + mhi;
            out[(size_t)mr * DTOT + ocol0]      = acc[0][t][v] + mv0;
            out[(size_t)mr * DTOT + ocol0 + 16] = acc[1][t][v] + mv1;
        }
    }
}

// ---------------------------------------------------------------------------
// z_z kernel: out[:, n] = m[n] + eps_z @ (tril(Lz)+jitter*I)^T,  n in [0,512)
// Grid: 4 blocks x 256 threads; wave owns one 16-column N-tile, all 8 M-tiles.
// Tiny (<1% of FLOPs); A read straight from L2-resident eps.
// ---------------------------------------------------------------------------
__global__ __launch_bounds__(256) void zz_wmma_kernel(
    const float* __restrict__ mvec,
    const float* __restrict__ Lz,        // (512,512)
    const float* __restrict__ eps,       // (128,66048)
    float* __restrict__ out)
{
    const int tid  = threadIdx.x;
    const int lane = tid & 31;
    const int wave = tid >> 5;
    const int ln   = lane & 15;
    const int hk   = (lane >> 4) << 1;

    const int n = blockIdx.x * 128 + wave * 16 + ln;   // output column 0..511

    v8f acc[8];
    #pragma unroll
    for (int t = 0; t < 8; ++t) acc[t] = v8f_zero();

    const float* __restrict__ bptr = Lz + (size_t)n * DZ + hk;
    for (int ks = 0; ks < DZ / 4; ++ks) {
        const int k = ks * 4 + hk;
        const v2f braw = *reinterpret_cast<const v2f*>(bptr + ks * 4);
        v2f b;
        b.x = (k     <= n) ? braw.x : 0.f; if (k     == n) b.x += JITTERF;
        b.y = (k + 1 <= n) ? braw.y : 0.f; if (k + 1 == n) b.y += JITTERF;
        #pragma unroll
        for (int t = 0; t < 8; ++t) {
            const v2f a = *reinterpret_cast<const v2f*>(
                eps + (size_t)(t * 16 + ln) * DTOT + k);
            acc[t] = wmma_k4(a, b, acc[t]);
        }
    }

    const float mval = mvec[n];
    const int mhi = (lane >> 4) << 3;
    #pragma unroll
    for (int t = 0; t < 8; ++t) {
        #pragma unroll
        for (int v = 0; v < 8; ++v) {
            const int mr = t * 16 + v + mhi;
            out[(size_t)mr * DTOT + n] = acc[t][v] + mval;
        }
    }
}

extern "C" void kernel_launch(void* const* d_in, const int* in_sizes, int n_in,
                              void* d_out, int out_size, void* d_ws, size_t ws_size,
                              hipStream_t stream) {
    const float* m   = (const float*)d_in[0];   // (66048)
    const float* Lz  = (const float*)d_in[1];   // (512,512)
    const float* Ly  = (const float*)d_in[2];   // (2048,32,32)
    const float* Lyz = (const float*)d_in[3];   // (65536,512)
    const float* eps = (const float*)d_in[4];   // (128,66048)
    float* out = (float*)d_out;                 // (128,66048)

    hipLaunchKernelGGL(zz_wmma_kernel, dim3(4),   dim3(256), 0, stream,
                       m, Lz, eps, out);
    hipLaunchKernelGGL(zy_wmma_kernel, dim3(512), dim3(256), 0, stream,
                       m, Ly, Lyz, eps, out);
}